// NormalEstimator_82970178224393
// MI455X (gfx1250) — compile-verified
//
#include <hip/hip_runtime.h>
#include <hip/hip_bf16.h>

// ---------------------------------------------------------------------------
// NormalEstimator for MI455X (gfx1250): KNN (radix-select) + WMMA-f32 MLP.
// All GEMMs use V_WMMA_F32_16X16X4_F32 (fp32 matrix pipe, full precision).
// Register-blocked: 32x32 tiles per wave (generic GEMM); A-tile hoisted to
// registers and reused across 8 column tiles (conv3 passes).
// ---------------------------------------------------------------------------

#define NPTS  100000
#define PQ    1000
#define KNB   50
#define KPAD  64          // pad K 50 -> 64 so row counts are multiples of 16
#define ROWS  (PQ * KPAD) // 64000 activation rows for the conv stack
#define FCROWS 1024       // pad 1000 -> 1024 rows for the FC stack
#define EPS   1e-5f

typedef __attribute__((ext_vector_type(2))) float v2f;
typedef __attribute__((ext_vector_type(8))) float v8f;

static __device__ __forceinline__ v8f wmma4(v2f a, v2f b, v8f c) {
  // D = A(16x4 f32) * B(4x16 f32) + C(16x16 f32)
  return __builtin_amdgcn_wmma_f32_16x16x4_f32(false, a, false, b,
                                               (short)0, c, false, false);
}

static __device__ __forceinline__ v8f zero8() {
  v8f z = {0.f, 0.f, 0.f, 0.f, 0.f, 0.f, 0.f, 0.f};
  return z;
}

// ---------------------------------------------------------------------------
// K0: per-query KNN via MSD radix select on L-inf distance bit patterns.
// Distances are non-negative floats -> bit pattern is monotone as uint32.
// Output: centered neighbor coords x[p][KPAD][3], rows k>=KNB zero-padded.
// ---------------------------------------------------------------------------
__global__ __launch_bounds__(256) void knn_kernel(
    const float* __restrict__ pts, const int* __restrict__ index,
    float* __restrict__ xout) {
  __shared__ unsigned hist[256];
  __shared__ unsigned s_byte, s_kth;
  __shared__ int sel[KNB];
  __shared__ int cntL, cntE;
  __shared__ float q[3];

  const int p = blockIdx.x;
  const int tid = threadIdx.x;

  if (tid == 0) {
    int qi = index[p];
    q[0] = pts[qi * 3 + 0];
    q[1] = pts[qi * 3 + 1];
    q[2] = pts[qi * 3 + 2];
  }
  __syncthreads();
  const float qx = q[0], qy = q[1], qz = q[2];

  unsigned prefix = 0;
  unsigned kth = KNB;  // rank we are hunting inside the current prefix group

  for (int shift = 24; shift >= 0; shift -= 8) {
    hist[tid] = 0u;
    __syncthreads();
    const unsigned hmask = (shift == 24) ? 0u : (0xFFFFFFFFu << (shift + 8));
    for (int i = tid; i < NPTS; i += 256) {
      float dx = fabsf(pts[i * 3 + 0] - qx);
      float dy = fabsf(pts[i * 3 + 1] - qy);
      float dz = fabsf(pts[i * 3 + 2] - qz);
      unsigned d = __float_as_uint(fmaxf(dx, fmaxf(dy, dz)));
      if ((d & hmask) == prefix)
        atomicAdd(&hist[(d >> shift) & 255u], 1u);
    }
    __syncthreads();
    if (tid == 0) {
      unsigned c = 0;
      for (unsigned b = 0; b < 256; ++b) {
        unsigned h = hist[b];
        if (c + h >= kth) { s_byte = b; s_kth = kth - c; break; }
        c += h;
      }
    }
    __syncthreads();
    prefix |= (s_byte << shift);
    kth = s_kth;
    __syncthreads();
  }

  const unsigned T = prefix;         // bit pattern of the K-th smallest dist
  const int needEq = (int)kth;       // how many dist==T elements we take
  const int cLess = KNB - needEq;    // number of elements strictly < T
  if (tid == 0) { cntL = 0; cntE = 0; }
  __syncthreads();

  for (int i = tid; i < NPTS; i += 256) {
    float dx = fabsf(pts[i * 3 + 0] - qx);
    float dy = fabsf(pts[i * 3 + 1] - qy);
    float dz = fabsf(pts[i * 3 + 2] - qz);
    unsigned d = __float_as_uint(fmaxf(dx, fmaxf(dy, dz)));
    if (d < T) {
      int s = atomicAdd(&cntL, 1);
      if (s < KNB) sel[s] = i;
    } else if (d == T) {
      int e = atomicAdd(&cntE, 1);
      int s = cLess + e;
      if (s < KNB) sel[s] = i;
    }
  }
  __syncthreads();

  for (int t = tid; t < KPAD; t += 256) {
    float vx = 0.f, vy = 0.f, vz = 0.f;
    if (t < KNB) {
      int i = sel[t];
      vx = pts[i * 3 + 0] - qx;
      vy = pts[i * 3 + 1] - qy;
      vz = pts[i * 3 + 2] - qz;
    }
    float* o = xout + ((size_t)p * KPAD + t) * 3;
    o[0] = vx; o[1] = vy; o[2] = vz;
  }
}

// ---------------------------------------------------------------------------
// Conv1: 3 -> 64, relu. Inner dim 3 -> plain VALU (too thin for WMMA).
// ---------------------------------------------------------------------------
__global__ __launch_bounds__(256) void conv1_kernel(
    const float* __restrict__ x, const float* __restrict__ W1,
    const float* __restrict__ b1, float* __restrict__ t1) {
  int row = blockIdx.x * blockDim.x + threadIdx.x;
  if (row >= ROWS) return;
  float x0 = x[row * 3 + 0], x1 = x[row * 3 + 1], x2 = x[row * 3 + 2];
  float* o = t1 + (size_t)row * 64;
  for (int c = 0; c < 64; ++c) {
    float v = W1[c * 3 + 0] * x0 + W1[c * 3 + 1] * x1 + W1[c * 3 + 2] * x2 + b1[c];
    o[c] = fmaxf(v, 0.f);
  }
}

// ---------------------------------------------------------------------------
// Generic WMMA GEMM, 32x32 register-blocked: Dst = relu(A @ W^T + bias).
// A  : [Mrows x Kdim] row-major (activations), Mrows % 32 == 0
// W  : [Ncols x Kdim] row-major (torch Linear layout), Ncols % 32 == 0
// One wave computes a 32x32 output tile (4 x v8f accumulators): per k-step
// 4 fragment loads feed 4 WMMAs (load:math 1:1 vs 2:1 for a 16x16 tile).
// A frag:  lane l holds A[m + (l&15), k + 2*(l>>4) + {0,1}]  (contig float2)
// C/D:     lane l, elem v -> row = 8*(l>>4)+v, col = l&15
// ---------------------------------------------------------------------------
__global__ __launch_bounds__(256) void gemm_relu_kernel(
    const float* __restrict__ A, const float* __restrict__ W,
    const float* __restrict__ bias, float* __restrict__ Dst,
    int Mrows, int Ncols, int Kdim) {
  const int wave = threadIdx.x >> 5;
  const int lane = threadIdx.x & 31;
  const int ntiles2 = Ncols >> 5;
  const int tile = blockIdx.x * (blockDim.x >> 5) + wave;
  const int mt2 = tile / ntiles2;
  const int nt2 = tile - mt2 * ntiles2;
  if (mt2 * 32 >= Mrows) return;

  const int r = lane & 15;
  const int kh = (lane >> 4) << 1;
  const float* __restrict__ A0 = A + (size_t)(mt2 * 32 + r) * Kdim + kh;
  const float* __restrict__ A1 = A0 + (size_t)16 * Kdim;
  const float* __restrict__ W0 = W + (size_t)(nt2 * 32 + r) * Kdim + kh;
  const float* __restrict__ W1 = W0 + (size_t)16 * Kdim;
  __builtin_prefetch(A0, 0, 1);
  __builtin_prefetch(W0, 0, 1);

  v8f acc00 = zero8(), acc01 = zero8(), acc10 = zero8(), acc11 = zero8();
  for (int k = 0; k < Kdim; k += 4) {
    v2f a0 = *(const v2f*)(A0 + k);
    v2f a1 = *(const v2f*)(A1 + k);
    v2f b0 = *(const v2f*)(W0 + k);
    v2f b1 = *(const v2f*)(W1 + k);
    acc00 = wmma4(a0, b0, acc00);
    acc01 = wmma4(a0, b1, acc01);
    acc10 = wmma4(a1, b0, acc10);
    acc11 = wmma4(a1, b1, acc11);
  }

  const int n0 = nt2 * 32 + r;
  const int n1 = n0 + 16;
  const float bs0 = bias[n0], bs1 = bias[n1];
  const int mb0 = mt2 * 32 + ((lane >> 4) << 3);
  const int mb1 = mb0 + 16;
#pragma unroll
  for (int v = 0; v < 8; ++v) {
    Dst[(size_t)(mb0 + v) * Ncols + n0] = fmaxf(acc00[v] + bs0, 0.f);
    Dst[(size_t)(mb0 + v) * Ncols + n1] = fmaxf(acc01[v] + bs1, 0.f);
    Dst[(size_t)(mb1 + v) * Ncols + n0] = fmaxf(acc10[v] + bs0, 0.f);
    Dst[(size_t)(mb1 + v) * Ncols + n1] = fmaxf(acc11[v] + bs1, 0.f);
  }
}

// ---------------------------------------------------------------------------
// Conv3 pass A (stats): h3 = relu(t2n @ W3^T + b3) on the fly; accumulate
// per-(p,channel) sum / sumsq over valid k<KNB rows. Values are discarded
// (recomputed in pass B) -- cheaper than spilling 205 MB to HBM.
// One block per p; wave owns 8 column tiles; the 16x128 A-tile is hoisted
// into 32 v2f registers per lane and reused across all 8 column tiles.
// ---------------------------------------------------------------------------
__global__ __launch_bounds__(256) void conv3_stats_kernel(
    const float* __restrict__ t2, const float* __restrict__ W3,
    const float* __restrict__ b3, float* __restrict__ part) {
  const int p = blockIdx.x;           // 0..999
  const int wave = threadIdx.x >> 5;
  const int lane = threadIdx.x & 31;
  const int r = lane & 15;
  const int kh = (lane >> 4) << 1;
  const float* __restrict__ Abase = t2 + (size_t)p * KPAD * 128;

  float s[8], sq[8];
#pragma unroll
  for (int ci = 0; ci < 8; ++ci) { s[ci] = 0.f; sq[ci] = 0.f; }

  for (int rt = 0; rt < 4; ++rt) {
    const float* __restrict__ Ap = Abase + (size_t)(rt * 16 + r) * 128 + kh;
    v2f areg[32];
#pragma unroll
    for (int kk = 0; kk < 32; ++kk) areg[kk] = *(const v2f*)(Ap + kk * 4);
    const int kbase = rt * 16 + ((lane >> 4) << 3);
#pragma unroll
    for (int ci = 0; ci < 8; ++ci) {
      const int n = (wave * 8 + ci) * 16 + r;
      const float* __restrict__ Wp = W3 + (size_t)n * 128 + kh;
      v8f acc = zero8();
#pragma unroll
      for (int kk = 0; kk < 32; ++kk)
        acc = wmma4(areg[kk], *(const v2f*)(Wp + kk * 4), acc);
      const float bs = b3[n];
#pragma unroll
      for (int v = 0; v < 8; ++v) {
        float val = fmaxf(acc[v] + bs, 0.f);
        if (kbase + v < KNB) { s[ci] += val; sq[ci] += val * val; }
      }
    }
  }
#pragma unroll
  for (int ci = 0; ci < 8; ++ci) {
    const int n = (wave * 8 + ci) * 16 + r;
    float ss = s[ci] + __shfl_xor(s[ci], 16);
    float qq = sq[ci] + __shfl_xor(sq[ci], 16);
    if (lane < 16) {
      part[((size_t)p * 1024 + n) * 2 + 0] = ss;
      part[((size_t)p * 1024 + n) * 2 + 1] = qq;
    }
  }
}

// ---------------------------------------------------------------------------
// Conv3 pass B: recompute h3, apply BN (scale/shift), max-pool over k<KNB.
// Same A-tile register-hoisted structure as pass A.
// feat rows p>=PQ are zero padding for the FC WMMA tiling.
// ---------------------------------------------------------------------------
__global__ __launch_bounds__(256) void conv3_max_kernel(
    const float* __restrict__ t2, const float* __restrict__ W3,
    const float* __restrict__ b3, const float* __restrict__ sc,
    const float* __restrict__ sh, float* __restrict__ feat) {
  const int p = blockIdx.x;           // 0..1023
  if (p >= PQ) {
    for (int c = threadIdx.x; c < 1024; c += blockDim.x)
      feat[(size_t)p * 1024 + c] = 0.f;
    return;
  }
  const int wave = threadIdx.x >> 5;
  const int lane = threadIdx.x & 31;
  const int r = lane & 15;
  const int kh = (lane >> 4) << 1;
  const float* __restrict__ Abase = t2 + (size_t)p * KPAD * 128;

  float mx[8];
#pragma unroll
  for (int ci = 0; ci < 8; ++ci) mx[ci] = -3.402823466e+38f;

  for (int rt = 0; rt < 4; ++rt) {
    const float* __restrict__ Ap = Abase + (size_t)(rt * 16 + r) * 128 + kh;
    v2f areg[32];
#pragma unroll
    for (int kk = 0; kk < 32; ++kk) areg[kk] = *(const v2f*)(Ap + kk * 4);
    const int kbase = rt * 16 + ((lane >> 4) << 3);
#pragma unroll
    for (int ci = 0; ci < 8; ++ci) {
      const int n = (wave * 8 + ci) * 16 + r;
      const float* __restrict__ Wp = W3 + (size_t)n * 128 + kh;
      v8f acc = zero8();
#pragma unroll
      for (int kk = 0; kk < 32; ++kk)
        acc = wmma4(areg[kk], *(const v2f*)(Wp + kk * 4), acc);
      const float bs = b3[n];
      const float scn = sc[n];
      const float shn = sh[n];
#pragma unroll
      for (int v = 0; v < 8; ++v) {
        float val = fmaxf(acc[v] + bs, 0.f) * scn + shn;
        if (kbase + v < KNB) mx[ci] = fmaxf(mx[ci], val);
      }
    }
  }
#pragma unroll
  for (int ci = 0; ci < 8; ++ci) {
    const int n = (wave * 8 + ci) * 16 + r;
    float m = fmaxf(mx[ci], __shfl_xor(mx[ci], 16));
    if (lane < 16) feat[(size_t)p * 1024 + n] = m;
  }
}

// ---------------------------------------------------------------------------
// BN stats on a stored tensor: fixed-order per-block partial sum/sumsq.
// mode 0: conv tensor, valid if (row & 63) < KNB ; mode 1: FC, valid if row<PQ
// ---------------------------------------------------------------------------
__global__ __launch_bounds__(256) void stats_partial_kernel(
    const float* __restrict__ t, float* __restrict__ part,
    int C, int rowsTotal, int rowsPerBlock, int mode) {
  const int b = blockIdx.x;
  int r0 = b * rowsPerBlock;
  int r1 = r0 + rowsPerBlock;
  if (r1 > rowsTotal) r1 = rowsTotal;
  for (int c = threadIdx.x; c < C; c += blockDim.x) {
    float s = 0.f, sq = 0.f;
    for (int r = r0; r < r1; ++r) {
      bool valid = (mode == 0) ? ((r & (KPAD - 1)) < KNB) : (r < PQ);
      if (valid) {
        float v = t[(size_t)r * C + c];
        s += v; sq += v * v;
      }
    }
    part[((size_t)b * C + c) * 2 + 0] = s;
    part[((size_t)b * C + c) * 2 + 1] = sq;
  }
}

// Reduce partials (fixed order) -> per-channel scale/shift for BN.
__global__ void finalize_stats_kernel(
    const float* __restrict__ part, int NB, int C, float invN,
    const float* __restrict__ g, const float* __restrict__ be,
    float* __restrict__ scale, float* __restrict__ shift) {
  int c = blockIdx.x * blockDim.x + threadIdx.x;
  if (c >= C) return;
  float s = 0.f, sq = 0.f;
  for (int b = 0; b < NB; ++b) {
    s += part[((size_t)b * C + c) * 2 + 0];
    sq += part[((size_t)b * C + c) * 2 + 1];
  }
  float m = s * invN;
  float v = sq * invN - m * m;
  float sc = g[c] * rsqrtf(v + EPS);
  scale[c] = sc;
  shift[c] = be[c] - m * sc;
}

// In-place per-channel affine: t = t*scale[c] + shift[c]. C is a power of 2.
__global__ __launch_bounds__(256) void normalize_kernel(
    float* __restrict__ t, const float* __restrict__ scale,
    const float* __restrict__ shift, int Cmask, int total) {
  int i = blockIdx.x * blockDim.x + threadIdx.x;
  if (i >= total) return;
  int c = i & Cmask;
  t[i] = t[i] * scale[c] + shift[c];
}

// Final 256 -> 3 layer (thin; scalar).
__global__ void fc3_out_kernel(const float* __restrict__ f2,
                               const float* __restrict__ Wf3,
                               const float* __restrict__ bf3,
                               float* __restrict__ out) {
  int t = blockIdx.x * blockDim.x + threadIdx.x;
  if (t >= PQ * 3) return;
  int p = t / 3, j = t - p * 3;
  float s = bf3[j];
  const float* fp = f2 + (size_t)p * 256;
  const float* wp = Wf3 + (size_t)j * 256;
  for (int c = 0; c < 256; ++c) s += fp[c] * wp[c];
  out[t] = s;
}

__global__ void copy_normals_kernel(const float* __restrict__ normal,
                                    const int* __restrict__ index,
                                    float* __restrict__ out) {
  int t = blockIdx.x * blockDim.x + threadIdx.x;
  if (t >= PQ * 3) return;
  int p = t / 3, j = t - p * 3;
  out[PQ * 3 + t] = normal[(size_t)index[p] * 3 + j];
}

// ---------------------------------------------------------------------------
extern "C" void kernel_launch(void* const* d_in, const int* in_sizes, int n_in,
                              void* d_out, int out_size, void* d_ws, size_t ws_size,
                              hipStream_t stream) {
  (void)in_sizes; (void)n_in; (void)out_size; (void)ws_size;
  const float* pts    = (const float*)d_in[0];
  const float* normal = (const float*)d_in[1];
  const int*   index  = (const int*)d_in[2];
  const float* W1  = (const float*)d_in[3];  const float* b1  = (const float*)d_in[4];
  const float* W2  = (const float*)d_in[5];  const float* b2  = (const float*)d_in[6];
  const float* W3  = (const float*)d_in[7];  const float* b3  = (const float*)d_in[8];
  const float* g1  = (const float*)d_in[9];  const float* be1 = (const float*)d_in[10];
  const float* g2  = (const float*)d_in[11]; const float* be2 = (const float*)d_in[12];
  const float* g3  = (const float*)d_in[13]; const float* be3 = (const float*)d_in[14];
  const float* Wf1 = (const float*)d_in[15]; const float* bf1 = (const float*)d_in[16];
  const float* g4  = (const float*)d_in[17]; const float* be4 = (const float*)d_in[18];
  const float* Wf2 = (const float*)d_in[19]; const float* bf2 = (const float*)d_in[20];
  const float* g5  = (const float*)d_in[21]; const float* be5 = (const float*)d_in[22];
  const float* Wf3 = (const float*)d_in[23]; const float* bf3 = (const float*)d_in[24];
  float* out = (float*)d_out;

  // Workspace carve-up (floats, 256-float aligned chunks).
  float* base = (float*)d_ws;
  size_t off = 0;
  auto alloc = [&](size_t n) {
    float* p = base + off;
    off += (n + 255) & ~(size_t)255;
    return p;
  };
  float* xbuf  = alloc((size_t)ROWS * 3);        // centered neighbors
  float* t1    = alloc((size_t)ROWS * 64);       // conv1 out (post-relu)
  float* t2    = alloc((size_t)ROWS * 128);      // conv2 out
  float* feat  = alloc((size_t)FCROWS * 1024);   // maxpooled features
  float* f1    = alloc((size_t)FCROWS * 512);
  float* f2    = alloc((size_t)FCROWS * 256);
  float* partA = alloc((size_t)250 * 64 * 2);
  float* partB = alloc((size_t)250 * 128 * 2);
  float* part3 = alloc((size_t)PQ * 1024 * 2);
  float* part4 = alloc((size_t)8 * 512 * 2);
  float* part5 = alloc((size_t)8 * 256 * 2);
  float* s1 = alloc(64);   float* h1 = alloc(64);
  float* s2 = alloc(128);  float* h2 = alloc(128);
  float* s3 = alloc(1024); float* h3 = alloc(1024);
  float* s4 = alloc(512);  float* h4 = alloc(512);
  float* s5 = alloc(256);  float* h5 = alloc(256);

  const float invConv = 1.f / (float)(PQ * KNB);  // 50000 valid elements
  const float invFC   = 1.f / (float)PQ;          // 1000 valid rows

  // ---- KNN + conv stack ----
  knn_kernel<<<PQ, 256, 0, stream>>>(pts, index, xbuf);
  conv1_kernel<<<ROWS / 256, 256, 0, stream>>>(xbuf, W1, b1, t1);

  stats_partial_kernel<<<250, 256, 0, stream>>>(t1, partA, 64, ROWS, 256, 0);
  finalize_stats_kernel<<<1, 64, 0, stream>>>(partA, 250, 64, invConv, g1, be1, s1, h1);
  normalize_kernel<<<(ROWS * 64) / 256, 256, 0, stream>>>(t1, s1, h1, 63, ROWS * 64);

  // 64000x128x64: (2000 * 4) 32x32 tiles / 8 waves = 1000 blocks
  gemm_relu_kernel<<<1000, 256, 0, stream>>>(t1, W2, b2, t2, ROWS, 128, 64);
  stats_partial_kernel<<<250, 256, 0, stream>>>(t2, partB, 128, ROWS, 256, 0);
  finalize_stats_kernel<<<1, 128, 0, stream>>>(partB, 250, 128, invConv, g2, be2, s2, h2);
  normalize_kernel<<<(ROWS * 128) / 256, 256, 0, stream>>>(t2, s2, h2, 127, ROWS * 128);

  conv3_stats_kernel<<<PQ, 256, 0, stream>>>(t2, W3, b3, part3);
  finalize_stats_kernel<<<4, 256, 0, stream>>>(part3, PQ, 1024, invConv, g3, be3, s3, h3);
  conv3_max_kernel<<<FCROWS, 256, 0, stream>>>(t2, W3, b3, s3, h3, feat);

  // ---- FC head ----
  // 1024x512x1024: (32 * 16) tiles / 8 = 64 blocks
  gemm_relu_kernel<<<64, 256, 0, stream>>>(feat, Wf1, bf1, f1, FCROWS, 512, 1024);
  stats_partial_kernel<<<8, 256, 0, stream>>>(f1, part4, 512, FCROWS, 128, 1);
  finalize_stats_kernel<<<2, 256, 0, stream>>>(part4, 8, 512, invFC, g4, be4, s4, h4);
  normalize_kernel<<<(FCROWS * 512) / 256, 256, 0, stream>>>(f1, s4, h4, 511, FCROWS * 512);

  // 1024x256x512: (32 * 8) tiles / 8 = 32 blocks
  gemm_relu_kernel<<<32, 256, 0, stream>>>(f1, Wf2, bf2, f2, FCROWS, 256, 512);
  stats_partial_kernel<<<8, 256, 0, stream>>>(f2, part5, 256, FCROWS, 128, 1);
  finalize_stats_kernel<<<1, 256, 0, stream>>>(part5, 8, 256, invFC, g5, be5, s5, h5);
  normalize_kernel<<<(FCROWS * 256) / 256, 256, 0, stream>>>(f2, s5, h5, 255, FCROWS * 256);

  fc3_out_kernel<<<12, 256, 0, stream>>>(f2, Wf3, bf3, out);
  copy_normals_kernel<<<12, 256, 0, stream>>>(normal, index, out);
}